// CellLoss_35270271434962
// MI455X (gfx1250) — compile-verified
//
#include <hip/hip_runtime.h>
#include <hip/hip_bf16.h>

// ---------------------------------------------------------------------------
// CellLoss: per-row gather + masked row reduction + mean.
//   B=131072 rows, C=256 cols f32. 128 MiB single-pass read -> pure HBM-bound
//   (~5.6us floor @ 23.3 TB/s). No GEMM -> no WMMA. gfx1250 levers used:
//   wave32 row ownership, async global->LDS double-buffered streaming
//   (ASYNCcnt), NT load hints, deterministic two-stage reduction (no atomics).
// ---------------------------------------------------------------------------

#if defined(__HIP_DEVICE_COMPILE__)
#if __has_builtin(__builtin_amdgcn_global_load_async_to_lds_b128) && \
    __has_builtin(__builtin_amdgcn_s_wait_asynccnt)
#define ASYNC_PATH 1
#endif
#endif
#ifndef ASYNC_PATH
#define ASYNC_PATH 0
#endif

#define WPB 8          // waves per block (256 threads, wave32)
#define MAXBLOCKS 2048

typedef float v4f __attribute__((ext_vector_type(4)));

// b128 async payload type exactly as the builtin expects: int vector_size(16),
// with explicit global (AS1) / LDS (AS3) qualification.
typedef int v4i_pay __attribute__((vector_size(16)));
typedef __attribute__((address_space(1))) v4i_pay* gas_b128_t;
typedef __attribute__((address_space(3))) v4i_pay* las_b128_t;

__device__ __forceinline__ float waveReduceAdd(float s) {
#pragma unroll
  for (int off = 16; off > 0; off >>= 1) s += __shfl_xor(s, off, 32);
  return s;  // all 32 lanes hold the total
}

// Direct-from-global row term (fallback / generic-C path). NT hints: one-pass.
__device__ __forceinline__ float rowTermDirect(const float* __restrict__ r,
                                               int C, int lbl, int lane) {
  float g = __builtin_nontemporal_load(r + lbl);  // same addr all lanes
  float s = 0.f;
  if ((C & 3) == 0) {
    const v4f* r4 = (const v4f*)r;
    for (int j = lane, n4 = C >> 2; j < n4; j += 32) {
      v4f v = __builtin_nontemporal_load(r4 + j);
      s += (v.x > g) ? v.x : 0.f;
      s += (v.y > g) ? v.y : 0.f;
      s += (v.z > g) ? v.z : 0.f;
      s += (v.w > g) ? v.w : 0.f;
    }
  } else {
    for (int j = lane; j < C; j += 32) {
      float v = __builtin_nontemporal_load(r + j);
      s += (v > g) ? v : 0.f;
    }
  }
  s = waveReduceAdd(s);
  return s + 1.0f / (g + 0.1f);
}

__device__ __forceinline__ void blockEmit(float acc, int lane, int w,
                                          float* __restrict__ ws) {
  __shared__ float wsum[WPB];
  if (lane == 0) wsum[w] = acc;
  __syncthreads();
  if (threadIdx.x == 0) {
    float t = 0.f;
#pragma unroll
    for (int i = 0; i < WPB; ++i) t += wsum[i];
    ws[blockIdx.x] = t;
  }
}

// ------------------- plain streaming kernel (generic C) --------------------
__global__ __launch_bounds__(256) void cellloss_partial_plain(
    const float* __restrict__ X, const int* __restrict__ L,
    float* __restrict__ ws, int B, int C) {
  const int lane = threadIdx.x & 31;
  const int w = threadIdx.x >> 5;
  const int gw = blockIdx.x * WPB + w;
  const int nw = gridDim.x * WPB;
  float acc = 0.f;
  for (int row = gw; row < B; row += nw)
    acc += rowTermDirect(X + (size_t)row * C, C, L[row], lane);
  blockEmit(acc, lane, w, ws);
}

// -------- async global->LDS double-buffered kernel (C % 128 == 0) ----------
// KPER = number of b128-per-lane copy steps per row = C/128 (immediate for
// s_wait_asynccnt). Each wave owns 2 row buffers in LDS; the next row's HBM
// fetch overlaps the current row's compare/reduce.
template <int KPER>
__global__ __launch_bounds__(256) void cellloss_partial_async(
    const float* __restrict__ X, const int* __restrict__ L,
    float* __restrict__ ws, int B, int C) {
  extern __shared__ float smem[];  // WPB * 2 * C floats
  const int lane = threadIdx.x & 31;
  const int w = threadIdx.x >> 5;
  const int gw = blockIdx.x * WPB + w;
  const int nw = gridDim.x * WPB;
  float acc = 0.f;

#if ASYNC_PATH
  float* buf0 = smem + (size_t)(2 * w) * C;
  float* buf1 = buf0 + C;
  int row = gw;
  if (row < B) {
    const char* src = (const char*)(X + (size_t)row * C);
    char* dst = (char*)buf0;
#pragma unroll
    for (int k = 0; k < KPER; ++k) {
      int off = lane * 16 + k * 512;
      __builtin_amdgcn_global_load_async_to_lds_b128(
          (gas_b128_t)(src + off), (las_b128_t)(dst + off), 0, 0);
    }
  }
  int p = 0;
  for (; row < B; row += nw) {
    const int nxt = row + nw;
    if (nxt < B) {  // prefetch next row into the other buffer, then wait
      const char* src = (const char*)(X + (size_t)nxt * C);
      char* dst = (char*)(p ? buf0 : buf1);
#pragma unroll
      for (int k = 0; k < KPER; ++k) {
        int off = lane * 16 + k * 512;
        __builtin_amdgcn_global_load_async_to_lds_b128(
            (gas_b128_t)(src + off), (las_b128_t)(dst + off), 0, 0);
      }
      __builtin_amdgcn_s_wait_asynccnt(KPER);  // oldest KPER (current row) done
    } else {
      __builtin_amdgcn_s_wait_asynccnt(0);
    }
    const float* cur = p ? buf1 : buf0;
    const int lbl = L[row];
    const float g = cur[lbl];
    float s = 0.f;
    const v4f* c4 = (const v4f*)cur;
    for (int j = lane, n4 = C >> 2; j < n4; j += 32) {
      v4f v = c4[j];
      s += (v.x > g) ? v.x : 0.f;
      s += (v.y > g) ? v.y : 0.f;
      s += (v.z > g) ? v.z : 0.f;
      s += (v.w > g) ? v.w : 0.f;
    }
    s = waveReduceAdd(s);
    acc += s + 1.0f / (g + 0.1f);
    p ^= 1;
  }
#else
  (void)smem;
  for (int row = gw; row < B; row += nw)
    acc += rowTermDirect(X + (size_t)row * C, C, L[row], lane);
#endif
  blockEmit(acc, lane, w, ws);
}

// ---------------------- final deterministic reduction ----------------------
__global__ __launch_bounds__(256) void cellloss_reduce(
    const float* __restrict__ ws, int n, float* __restrict__ out, float invB) {
  __shared__ float sm[256];
  float s = 0.f;
  for (int i = threadIdx.x; i < n; i += 256) s += ws[i];
  sm[threadIdx.x] = s;
  __syncthreads();
#pragma unroll
  for (int k = 128; k > 0; k >>= 1) {
    if ((int)threadIdx.x < k) sm[threadIdx.x] += sm[threadIdx.x + k];
    __syncthreads();
  }
  if (threadIdx.x == 0) out[0] = sm[0] * invB;
}

extern "C" void kernel_launch(void* const* d_in, const int* in_sizes, int n_in,
                              void* d_out, int out_size, void* d_ws,
                              size_t ws_size, hipStream_t stream) {
  const float* X = (const float*)d_in[0];
  const int* L = (const int*)d_in[1];
  float* out = (float*)d_out;
  float* ws = (float*)d_ws;

  const int B = in_sizes[1];
  const int C = in_sizes[0] / (B > 0 ? B : 1);

  int grid = (B + WPB - 1) / WPB;
  if (grid > MAXBLOCKS) grid = MAXBLOCKS;
  if (grid < 1) grid = 1;
  const int maxg = (int)(ws_size / sizeof(float));
  if (maxg > 0 && grid > maxg) grid = maxg;

  bool launched = false;
  if ((C % 128) == 0 && C > 0) {
    const int kper = C / 128;
    const size_t dynBytes = (size_t)WPB * 2 * (size_t)C * sizeof(float);
    if (dynBytes <= 96 * 1024) {
      switch (kper) {
        case 1:
          cellloss_partial_async<1><<<grid, 256, dynBytes, stream>>>(X, L, ws, B, C);
          launched = true;
          break;
        case 2:
          cellloss_partial_async<2><<<grid, 256, dynBytes, stream>>>(X, L, ws, B, C);
          launched = true;
          break;
        case 3:
          cellloss_partial_async<3><<<grid, 256, dynBytes, stream>>>(X, L, ws, B, C);
          launched = true;
          break;
        case 4:
          cellloss_partial_async<4><<<grid, 256, dynBytes, stream>>>(X, L, ws, B, C);
          launched = true;
          break;
        default:
          break;
      }
    }
  }
  if (!launched)
    cellloss_partial_plain<<<grid, 256, 0, stream>>>(X, L, ws, B, C);

  cellloss_reduce<<<1, 256, 0, stream>>>(ws, grid, out, 1.0f / (float)B);
}